// EdgeEnhancedGNN_15229954031644
// MI455X (gfx1250) — compile-verified
//
#include <hip/hip_runtime.h>
#include <hip/hip_bf16.h>

#define N_NODES   100000
#define N_EDGES   1600000
#define HIDDEN    128
#define N_GRAPHS  1024
#define NODE_DIM  13
#define HID2      64     // HIDDEN/2
#define LDS_PITCH 132    // HIDDEN + 4 words: conflict-free b64 A-fragment reads

typedef __attribute__((ext_vector_type(2))) float v2f;
typedef __attribute__((ext_vector_type(8))) float v8f;

// ---------------------------------------------------------------------------
// Kernel 1: h = x @ Wn + bn        [N,13] @ [13,128]
// K=13 is too small/odd for WMMA; one thread per (node, col), Wn hits L2/L0.
// ---------------------------------------------------------------------------
__global__ void k_node_embed(const float* __restrict__ x,
                             const float* __restrict__ Wn,
                             const float* __restrict__ bn,
                             float* __restrict__ h) {
    int idx = blockIdx.x * blockDim.x + threadIdx.x;
    if (idx >= N_NODES * HIDDEN) return;
    int n = idx >> 7;
    int c = idx & (HIDDEN - 1);
    const float* xr = x + n * NODE_DIM;
    float acc = bn[c];
#pragma unroll
    for (int k = 0; k < NODE_DIM; ++k)
        acc = fmaf(xr[k], Wn[k * HIDDEN + c], acc);
    h[idx] = acc;
}

// ---------------------------------------------------------------------------
// Kernel 2 (per layer): msg = relu(h[src] + dist*We + be); scatter-add into
// aggr[dst]. Persistent waves; each wave sweeps edges in chunks of 32:
//   - We4/be4 hoisted out of the loop (constant per lane)
//   - src/dst/dist loaded COALESCED (one edge per lane), then broadcast
//     lane-by-lane via __shfl (wave32 permute path)
//   - each lane then gathers 4 channels of h[src] (float4) and issues 4
//     global_atomic_add_f32 into aggr[dst]; h/aggr are L2-resident (51 MB).
// N_EDGES is a multiple of 32, so chunks are always full.
// ---------------------------------------------------------------------------
__global__ void k_edge_msg(const float* __restrict__ h,
                           const float* __restrict__ edge_dist,
                           const int*   __restrict__ src,
                           const int*   __restrict__ dst,
                           const float* __restrict__ We,
                           const float* __restrict__ be,
                           float* __restrict__ aggr) {
    int gid    = blockIdx.x * blockDim.x + threadIdx.x;
    int wave   = gid >> 5;
    int lane   = threadIdx.x & 31;
    int nwaves = (gridDim.x * blockDim.x) >> 5;
    int c0     = lane * 4;

    const float4 wv = *(const float4*)(We + c0);   // loop-invariant
    const float4 bv = *(const float4*)(be + c0);

    for (int base = wave * 32; base < N_EDGES; base += nwaves * 32) {
        int   e  = base + lane;
        int   sL = src[e];          // coalesced: 32 consecutive edges
        int   dL = dst[e];
        float tL = edge_dist[e];

#pragma unroll 4
        for (int j = 0; j < 32; ++j) {
            int   sj = __shfl(sL, j, 32);
            int   dj = __shfl(dL, j, 32);
            float tj = __shfl(tL, j, 32);

            float4 hv = *(const float4*)(h + (size_t)sj * HIDDEN + c0);
            float m0 = fmaxf(fmaf(tj, wv.x, bv.x) + hv.x, 0.0f);
            float m1 = fmaxf(fmaf(tj, wv.y, bv.y) + hv.y, 0.0f);
            float m2 = fmaxf(fmaf(tj, wv.z, bv.z) + hv.z, 0.0f);
            float m3 = fmaxf(fmaf(tj, wv.w, bv.w) + hv.w, 0.0f);

            float* ar = aggr + (size_t)dj * HIDDEN + c0;
            atomicAdd(ar + 0, m0);
            atomicAdd(ar + 1, m1);
            atomicAdd(ar + 2, m2);
            atomicAdd(ar + 3, m3);
        }
    }
}

// ---------------------------------------------------------------------------
// Kernel 3 (per layer): fused GINE combine + conv GEMM + relu + residual,
// using V_WMMA_F32_16X16X4_F32 (exact fp32 — matches the fp32 reference).
//   upd  = (1+eps)*h + aggr      (computed ONCE per block into LDS)
//   hout = relu(upd @ W + bias) + h
// Block = 256 threads = 8 waves; wave w owns the 16-wide column tile w;
// all waves share the LDS-staged 16x128 A tile (row pitch 132 words ->
// the 32-lane ds_load_b64 A-fragment fetch covers all 64 banks exactly once).
//
// VGPR layouts per ISA 7.12.2 (32-bit, wave32):
//   A 16x4 : lane M = lane&15 ; lanes<16 hold K={0,1}, lanes>=16 hold K={2,3}
//   B 4x16 : lane N = lane&15 ; same K split across lane halves
//   C/D    : VGPR r -> M=r (lanes 0-15) / M=r+8 (lanes 16-31), N = lane&15
// ---------------------------------------------------------------------------
__global__ void k_gine_conv(const float* __restrict__ h,
                            const float* __restrict__ aggr,
                            const float* __restrict__ W,     // [128,128] this layer
                            const float* __restrict__ bias,  // [128]     this layer
                            const float* __restrict__ eps,   // [3]
                            int layer,
                            float* __restrict__ hout) {
    __shared__ float updS[16 * LDS_PITCH];      // 8448 B

    int tid  = threadIdx.x;
    int lane = tid & 31;
    int wavE = tid >> 5;                        // 0..7 -> column tile
    int row0 = blockIdx.x * 16;
    int col0 = wavE * 16;

    float epsl = 1.0f + eps[layer];

    // Cooperative: upd tile -> LDS (each thread does 8 elements, coalesced)
#pragma unroll
    for (int i = tid; i < 16 * HIDDEN; i += 256) {
        int r = i >> 7;
        int k = i & (HIDDEN - 1);
        size_t g = (size_t)(row0 + r) * HIDDEN + k;
        updS[r * LDS_PITCH + k] = fmaf(epsl, h[g], aggr[g]);
    }
    __syncthreads();

    int mn    = lane & 15;                      // M for A, N for B / D-store
    int khalf = (lane >> 4) * 2;                // 0 or 2
    const float* As = updS + mn * LDS_PITCH;

    v8f c = {0.f, 0.f, 0.f, 0.f, 0.f, 0.f, 0.f, 0.f};

#pragma unroll 4
    for (int k0 = 0; k0 < HIDDEN; k0 += 4) {
        int ka = k0 + khalf;
        v2f a = *(const v2f*)(As + ka);         // ds_load_b64, conflict-free
        v2f b;
        b.x = W[(ka)     * HIDDEN + col0 + mn];
        b.y = W[(ka + 1) * HIDDEN + col0 + mn];
        c = __builtin_amdgcn_wmma_f32_16x16x4_f32(
                /*neg_a=*/false, a, /*neg_b=*/false, b,
                /*c_mod=*/(short)0, c, /*reuse_a=*/false, /*reuse_b=*/false);
    }

    int   mbase = (lane >> 4) * 8;              // 0 or 8
    float bcol  = bias[col0 + mn];
#pragma unroll
    for (int r = 0; r < 8; ++r) {
        int row = row0 + mbase + r;
        int col = col0 + mn;
        float v = fmaxf(c[r] + bcol, 0.0f) + h[(size_t)row * HIDDEN + col];
        hout[(size_t)row * HIDDEN + col] = v;
    }
}

// ---------------------------------------------------------------------------
// Kernel 4: segment-sum pooling accumulators. One wave32 per node; lane 0
// bumps the per-graph count. sums/cnt zeroed by host-side memset each call.
// ---------------------------------------------------------------------------
__global__ void k_pool_accum(const float* __restrict__ h,
                             const int*   __restrict__ batch,
                             float* __restrict__ sums,
                             float* __restrict__ cnt) {
    int gid  = blockIdx.x * blockDim.x + threadIdx.x;
    int node = gid >> 5;
    if (node >= N_NODES) return;
    int lane = gid & 31;
    int b  = batch[node];
    int c0 = lane * 4;
    float4 hv = *(const float4*)(h + (size_t)node * HIDDEN + c0);
    float* sr = sums + (size_t)b * HIDDEN + c0;
    atomicAdd(sr + 0, hv.x);
    atomicAdd(sr + 1, hv.y);
    atomicAdd(sr + 2, hv.z);
    atomicAdd(sr + 3, hv.w);
    if (lane == 0) atomicAdd(&cnt[b], 1.0f);
}

// ---------------------------------------------------------------------------
// Kernel 5: mean + MLP head. One block (128 thr) per graph.
//   pooled = sums/max(cnt,1); out = relu(pooled@Wd1+bd1) @ Wd2 + bd2
// ---------------------------------------------------------------------------
__global__ void k_head(const float* __restrict__ sums,
                       const float* __restrict__ cnt,
                       const float* __restrict__ Wd1,  // [128,64]
                       const float* __restrict__ bd1,  // [64]
                       const float* __restrict__ Wd2,  // [64,1]
                       const float* __restrict__ bd2,  // [1]
                       float* __restrict__ out) {
    __shared__ float pooled[HIDDEN];
    __shared__ float hid[HID2];
    int g = blockIdx.x;
    int t = threadIdx.x;

    float inv = 1.0f / fmaxf(cnt[g], 1.0f);
    if (t < HIDDEN) pooled[t] = sums[(size_t)g * HIDDEN + t] * inv;
    __syncthreads();

    if (t < HID2) {
        float acc = bd1[t];
#pragma unroll 8
        for (int k = 0; k < HIDDEN; ++k)
            acc = fmaf(pooled[k], Wd1[k * HID2 + t], acc);
        hid[t] = fmaxf(acc, 0.0f) * Wd2[t];
    }
    __syncthreads();
#pragma unroll
    for (int s = HID2 / 2; s > 0; s >>= 1) {
        if (t < s) hid[t] += hid[t + s];
        __syncthreads();
    }
    if (t == 0) out[g] = hid[0] + bd2[0];
}

// ---------------------------------------------------------------------------
extern "C" void kernel_launch(void* const* d_in, const int* in_sizes, int n_in,
                              void* d_out, int out_size, void* d_ws, size_t ws_size,
                              hipStream_t stream) {
    const float* x         = (const float*)d_in[0];
    const float* edge_dist = (const float*)d_in[1];
    const int*   edge_idx  = (const int*)  d_in[2];   // [2, E]
    const int*   batch     = (const int*)  d_in[3];
    const float* Wn        = (const float*)d_in[4];
    const float* bn        = (const float*)d_in[5];
    const float* We        = (const float*)d_in[6];
    const float* be        = (const float*)d_in[7];
    const float* convW     = (const float*)d_in[8];   // [3,128,128]
    const float* convb     = (const float*)d_in[9];   // [3,128]
    const float* eps       = (const float*)d_in[10];  // [3]
    const float* Wd1       = (const float*)d_in[11];
    const float* bd1       = (const float*)d_in[12];
    const float* Wd2       = (const float*)d_in[13];
    const float* bd2       = (const float*)d_in[14];
    float* out = (float*)d_out;

    const int* src = edge_idx;            // edge_index[0]
    const int* dst = edge_idx + N_EDGES;  // edge_index[1]

    // Workspace layout (floats): hA | hB | aggr | sums | cnt
    const size_t NH = (size_t)N_NODES * HIDDEN;
    float* hA   = (float*)d_ws;
    float* hB   = hA + NH;
    float* aggr = hB + NH;
    float* sums = aggr + NH;
    float* cnt  = sums + (size_t)N_GRAPHS * HIDDEN;

    // h0 = x @ Wn + bn
    {
        int total = N_NODES * HIDDEN;
        k_node_embed<<<(total + 255) / 256, 256, 0, stream>>>(x, Wn, bn, hA);
    }

    // 3 GINE layers, double-buffered hA <-> hB
    float* hin  = hA;
    float* hout = hB;
    for (int l = 0; l < 3; ++l) {
        hipMemsetAsync(aggr, 0, NH * sizeof(float), stream);

        // persistent-wave edge sweep: 4096 blocks x 256 thr = 32768 waves
        k_edge_msg<<<4096, 256, 0, stream>>>(
            hin, edge_dist, src, dst, We, be, aggr);

        k_gine_conv<<<N_NODES / 16, 256, 0, stream>>>(
            hin, aggr, convW + (size_t)l * HIDDEN * HIDDEN, convb + l * HIDDEN,
            eps, l, hout);

        float* t = hin; hin = hout; hout = t;
    }
    // final h is in `hin` (hA->hB->hA->hB, 3 swaps => hin == hB)

    // global mean pool + head
    hipMemsetAsync(sums, 0, ((size_t)N_GRAPHS * HIDDEN + N_GRAPHS) * sizeof(float), stream);
    {
        long long pthreads = (long long)N_NODES * 32;
        k_pool_accum<<<(int)((pthreads + 255) / 256), 256, 0, stream>>>(
            hin, batch, sums, cnt);
    }
    k_head<<<N_GRAPHS, 128, 0, stream>>>(sums, cnt, Wd1, bd1, Wd2, bd2, out);
}